// InterpLinear_33569464385943
// MI455X (gfx1250) — compile-verified
//
#include <hip/hip_runtime.h>
#include <cstdint>

typedef float v4f __attribute__((ext_vector_type(4)));
typedef int   v4i __attribute__((ext_vector_type(4)));

constexpr int Bq = 8, Tq = 128, Dq = 64, Kq = 3;
constexpr int NROWS = Tq * Kq;   // 384 (j,k) rows per (b,i) block

__global__ __launch_bounds__(256) void gl_interp_kernel(
    const float* __restrict__ x, const float* __restrict__ t,
    float* __restrict__ out)
{
  __shared__ v4f   sx[Tq * (Dq / 4)];   // 32 KB: x[b] as float4 rows of 16
  __shared__ float st[Tq];              // 512 B: t[b]
  __shared__ v4i   srow[NROWS];         // 6 KB: per-row {idx, w, scale, -}

  const int tid = threadIdx.x;
  const int b = blockIdx.x / Tq;
  const int i = blockIdx.x - b * Tq;

  const float* xg = x + (size_t)b * (Tq * Dq);

  // ---- Stage x[b] into LDS via async b128 loads (ASYNCcnt path) ----
  // 2048 float4 chunks, 8 per thread. Low 32 bits of a generic shared
  // pointer are the LDS byte offset (LDS aperture truncation).
#pragma unroll
  for (int p = 0; p < 8; ++p) {
    const int c = tid + p * 256;
    const uint32_t loff = (uint32_t)(uintptr_t)&sx[c];
    const float* gp = xg + (size_t)c * 4;
    asm volatile("global_load_async_to_lds_b128 %0, %1, off"
                 :: "v"(loff), "v"(gp)
                 : "memory");
  }
  if (tid < Tq) st[tid] = t[b * Tq + tid];
  asm volatile("s_wait_asynccnt 0" ::: "memory");
  __syncthreads();   // st + sx visible to all waves

  // ---- Phase 1: resolve all 384 rows' (idx, w, scale) once per block ----
  const float t_i = st[i];
#pragma unroll
  for (int rr = 0; rr < 2; ++rr) {
    const int r = tid + rr * 256;
    if (r < NROWS) {
      const int j = r / 3;
      const int k = r - j * 3;
      const float node = (k == 0) ? -0.77459f : ((k == 1) ? 0.0f : 0.77459f);
      const float glw  = (k == 1) ? 0.88888f : 0.55555f;

      const float t_j = st[j];
      float s;
      {
#pragma clang fp contract(off)
        const float itp = t_j + (t_i - t_j) * ((node + 1.0f) * 0.5f);
        s = floorf(itp);  // ATOL = 1.0
      }

      // searchsorted(t[b], s, side='right') - 1, clipped to [0, T-2]
      int lo = 0, hi = Tq;
#pragma unroll
      for (int stp = 0; stp < 7; ++stp) {
        const int mid = (lo + hi) >> 1;
        if (st[mid] <= s) lo = mid + 1; else hi = mid;
      }
      int idx = lo - 1;
      idx = idx < 0 ? 0 : (idx > Tq - 2 ? Tq - 2 : idx);

      const float tL = st[idx];
      const float tR = st[idx + 1];
      float w;
      {
#pragma clang fp contract(off)
        w = (s - tL) / (tR - tL);
      }
      w = w < 0.0f ? 0.0f : (w > 1.0f ? 1.0f : w);

      v4i pk;
      pk.x = idx;
      pk.y = __float_as_int(w);
      pk.z = __float_as_int(glw * 0.5f);  // exact (x * 2^-1)
      pk.w = 0;
      srow[r] = pk;
    }
  }
  __syncthreads();   // srow visible

  // ---- Main loop: pure gather + blend + streaming store ----
  const int wave = tid >> 5;
  const int lane = tid & 31;
  const int half = (lane >> 4) & 1;   // which of 2 rows this wave handles
  const int sub  = lane & 15;         // float4 index within the D=64 row

  // out[b,i,j,k,d] flattens to base + r*64 + d  (r = j*3 + k)
  float* ob = out + ((size_t)b * Tq + i) * (size_t)(Tq * Kq * Dq);

#pragma unroll 4
  for (int itr = 0; itr < NROWS / 16; ++itr) {
    const int r = itr * 16 + wave * 2 + half;   // 0..383

    const v4i pk = srow[r];                     // broadcast b128 within half-wave
    const int idx = pk.x;
    const float w     = __int_as_float(pk.y);
    const float scale = __int_as_float(pk.z);

    const v4f xL = sx[idx * 16 + sub];
    const v4f xR = sx[idx * 16 + 16 + sub];
    v4f o;
    {
#pragma clang fp contract(off)
      o = (xL + w * (xR - xL)) * scale;
    }

    // Streaming output is never re-read: non-temporal b128 store.
    __builtin_nontemporal_store(o, (v4f*)(ob + (size_t)r * Dq + sub * 4));
  }
}

extern "C" void kernel_launch(void* const* d_in, const int* in_sizes, int n_in,
                              void* d_out, int out_size, void* d_ws, size_t ws_size,
                              hipStream_t stream) {
  (void)in_sizes; (void)n_in; (void)d_ws; (void)ws_size; (void)out_size;
  const float* x = (const float*)d_in[0];
  const float* t = (const float*)d_in[1];
  float* out = (float*)d_out;
  dim3 grid(Bq * Tq), block(256);
  hipLaunchKernelGGL(gl_interp_kernel, grid, block, 0, stream, x, t, out);
}